// FSIPM_45681272160564
// MI455X (gfx1250) — compile-verified
//
#include <hip/hip_runtime.h>
#include <hip/hip_bf16.h>

// Problem constants
#define B_     64
#define ROIS_  400
#define T_     1200
#define C1_    8
#define C2_    16
#define HID_   100
#define SLOPE_ 0.01f

// Gram/WMMA tiling
#define MT_     64     // macro-tile edge (4 waves x 32x32)
#define NBLK_   7      // ceil(448/64); 400 padded to 448
#define NPAIR_  28     // NBLK_*(NBLK_+1)/2 upper-triangular macro blocks
#define KC_     60     // K chunk (1200 = 20*60)
#define LDK_    68     // padded LDS row stride (dwords) -> conflict-free b64 frag loads
#define NCHUNK_ 20
#define BUFSZ_  (2 * MT_ * LDK_)   // floats per double-buffer slot (A rows | B rows)

typedef __attribute__((ext_vector_type(2))) float v2f;
typedef __attribute__((ext_vector_type(8))) float v8f;
typedef int v4i_ __attribute__((vector_size(16)));   // matches builtin param type

__device__ __forceinline__ float leakyf(float v) { return v >= 0.f ? v : SLOPE_ * v; }

// ---- CDNA5 async global->LDS copy path (ASYNCcnt) --------------------------
#if __has_builtin(__builtin_amdgcn_global_load_async_to_lds_b128)
#define FSIPM_ASYNC 1
#define ASYNC_CP16(gsrc, ldst)                                                  \
  __builtin_amdgcn_global_load_async_to_lds_b128(                               \
      (__attribute__((address_space(1))) v4i_*)(gsrc),                          \
      (__attribute__((address_space(3))) v4i_*)(ldst), 0, 0)
#if __has_builtin(__builtin_amdgcn_s_wait_asynccnt)
#define WAIT_ASYNC0() __builtin_amdgcn_s_wait_asynccnt(0)
#else
#define WAIT_ASYNC0() asm volatile("s_wait_asynccnt 0x0" ::: "memory")
#endif
#else
#define FSIPM_ASYNC 0
#endif

// ---------------------------------------------------------------------------
// Kernel 1: per-(b,r) mean and inverse centered norm.
//   mean = (1/T) sum x ;  var = sum x^2 - T*mean^2 ;  invn = rsqrt(var)
// Centering folds into the Gram epilogue: Gram(xc) = Gram(x) - T*mu_r*mu_i
// ---------------------------------------------------------------------------
__global__ __launch_bounds__(64)
void fsipm_stats_kernel(const float* __restrict__ x,
                        float* __restrict__ MU, float* __restrict__ INVN) {
  const int r = blockIdx.x, b = blockIdx.y, tid = threadIdx.x;
  const float* row = x + ((size_t)b * ROIS_ + r) * T_;
  float s = 0.f, s2 = 0.f;
  for (int t = tid; t < T_; t += 64) { float v = row[t]; s += v; s2 += v * v; }
  __shared__ float rs[64], rs2[64];
  rs[tid] = s; rs2[tid] = s2;
  __syncthreads();
  for (int off = 32; off > 0; off >>= 1) {
    if (tid < off) { rs[tid] += rs[tid + off]; rs2[tid] += rs2[tid + off]; }
    __syncthreads();
  }
  if (tid == 0) {
    float mean = rs[0] * (1.0f / T_);
    float var = rs2[0] - (float)T_ * mean * mean;
    MU[b * ROIS_ + r] = mean;
    INVN[b * ROIS_ + r] = rsqrtf(fmaxf(var, 1e-30f));
  }
}

// ---------------------------------------------------------------------------
// Kernel 2: batched Gram via V_WMMA_F32_16X16X4_F32 + fused pcc epilogue.
// Upper-triangular 64x64 macro blocks only; mirror-store exploits symmetry.
// 4 waves/block, each wave owns a 32x32 sub-tile (4 accumulators).
// Staging: double-buffered GLOBAL_LOAD_ASYNC_TO_LDS_B128 (copy overlaps WMMA)
// when available; VGPR+ds_store fallback otherwise.
// ---------------------------------------------------------------------------
__global__ __launch_bounds__(128)
void fsipm_gram_pcc_kernel(const float* __restrict__ x,
                           const float* __restrict__ MU,
                           const float* __restrict__ INVN,
                           float* __restrict__ PCC) {
#if FSIPM_ASYNC
  __shared__ float sm[2 * BUFSZ_];  // 69.6 KB, two chunk buffers
#else
  __shared__ float sm[BUFSZ_];      // 34.8 KB, single buffer
#endif

  const int b = blockIdx.y;
  // decode upper-triangular (rowBlk, colBlk) pair from blockIdx.x
  int pr = blockIdx.x, rowBlk = 0;
  while (pr >= NBLK_ - rowBlk) { pr -= NBLK_ - rowBlk; ++rowBlk; }
  const int colBlk = rowBlk + pr;
  const int rowBase = rowBlk * MT_, colBase = colBlk * MT_;

  const int tid = threadIdx.x;
  const int lane = tid & 31;
  const int wid = tid >> 5;
  const int waveRow = wid >> 1, waveCol = wid & 1;

  // Fragment addressing per ISA 16x16x4 f32 layout:
  //   A: lane L holds row m=L&15, K = k + 2*(L>>4) + {0,1}  (2 consecutive f32)
  //   B symmetric: lane L holds col n=L&15, same K pair.
  const int m = lane & 15, half = lane >> 4;
  const int aOffBase = (waveRow * 32 + m) * LDK_ + 2 * half;
  const int bOffBase = MT_ * LDK_ + (waveCol * 32 + m) * LDK_ + 2 * half;

  v8f acc00 = {}, acc01 = {}, acc10 = {}, acc11 = {};

  const size_t xb = (size_t)b * ROIS_ * T_;

  // one K-chunk of WMMA work on an LDS buffer
  auto computeChunk = [&](const float* s) {
#pragma unroll
    for (int k = 0; k < KC_; k += 4) {
      v2f a0 = *(const v2f*)(&s[aOffBase + k]);
      v2f a1 = *(const v2f*)(&s[aOffBase + 16 * LDK_ + k]);
      v2f b0 = *(const v2f*)(&s[bOffBase + k]);
      v2f b1 = *(const v2f*)(&s[bOffBase + 16 * LDK_ + k]);
      acc00 = __builtin_amdgcn_wmma_f32_16x16x4_f32(false, a0, false, b0, (short)0, acc00, false, false);
      acc01 = __builtin_amdgcn_wmma_f32_16x16x4_f32(false, a0, false, b1, (short)0, acc01, false, false);
      acc10 = __builtin_amdgcn_wmma_f32_16x16x4_f32(false, a1, false, b0, (short)0, acc10, false, false);
      acc11 = __builtin_amdgcn_wmma_f32_16x16x4_f32(false, a1, false, b1, (short)0, acc11, false, false);
    }
  };

#if FSIPM_ASYNC
  // issue all async copies for one chunk into buffer slot `slot`
  auto issueChunk = [&](int chunk, int slot) {
    const int kc = chunk * KC_;
    float* base = sm + slot * BUFSZ_;
    for (int idx = tid; idx < 2 * MT_ * 15; idx += 128) {
      const int buf = idx / (MT_ * 15);
      const int rem = idx - buf * (MT_ * 15);
      const int row = rem / 15;
      const int col = rem - row * 15;
      const int gr = (buf ? colBase : rowBase) + row;
      float* dst = base + buf * MT_ * LDK_ + row * LDK_ + col * 4;
      if (gr < ROIS_) {
        const float* src = x + xb + (size_t)gr * T_ + kc + col * 4;
        ASYNC_CP16(src, dst);
      } else {
        *(float4*)dst = make_float4(0.f, 0.f, 0.f, 0.f);  // zero-pad rows >= 400
      }
    }
  };

  issueChunk(0, 0);
  WAIT_ASYNC0();
  __syncthreads();
  for (int c = 0; c < NCHUNK_; ++c) {
    const int cur = c & 1;
    if (c + 1 < NCHUNK_) issueChunk(c + 1, cur ^ 1);  // overlaps compute below
    computeChunk(sm + cur * BUFSZ_);
    if (c + 1 < NCHUNK_) {
      WAIT_ASYNC0();     // my next-chunk copies landed
      __syncthreads();   // everyone's landed; everyone done reading `cur`
    }
  }
#else
  for (int chunk = 0; chunk < NCHUNK_; ++chunk) {
    const int kc = chunk * KC_;
    __syncthreads();
    for (int idx = tid; idx < 2 * MT_ * 15; idx += 128) {
      const int buf = idx / (MT_ * 15);
      const int rem = idx - buf * (MT_ * 15);
      const int row = rem / 15;
      const int col = rem - row * 15;
      const int gr = (buf ? colBase : rowBase) + row;
      float4 v = make_float4(0.f, 0.f, 0.f, 0.f);
      if (gr < ROIS_) {
        const float4* src = (const float4*)(x + xb + (size_t)gr * T_ + kc);
        v = src[col];
      }
      *(float4*)(&sm[buf * MT_ * LDK_ + row * LDK_ + col * 4]) = v;
    }
    __syncthreads();
    computeChunk(sm);
  }
#endif

  // Epilogue: center-correct, normalize to pcc, mirror-store (symmetry).
  const float* mu = MU + b * ROIS_;
  const float* inv = INVN + b * ROIS_;
  float* pccB = PCC + (size_t)b * ROIS_ * ROIS_;
  const int rT0 = rowBase + waveRow * 32;
  const int cT0 = colBase + waveCol * 32;
  const int N = lane & 15;
  v8f accs[4] = {acc00, acc01, acc10, acc11};
#pragma unroll
  for (int t = 0; t < 4; ++t) {
    const int rt = rT0 + ((t >> 1) ? 16 : 0);
    const int ct = cT0 + ((t & 1) ? 16 : 0);
    const int ig = ct + N;
    if (ig < ROIS_) {
      const float mui = mu[ig], ini = inv[ig];
#pragma unroll
      for (int j = 0; j < 8; ++j) {
        const int rg = rt + j + 8 * half;  // C/D layout: M = j + 8*(lane>>4)
        if (rg < ROIS_) {
          const float mur = mu[rg], inr = inv[rg];
          const float v = (accs[t][j] - (float)T_ * mur * mui) * inr * ini;
          pccB[(size_t)rg * ROIS_ + ig] = v;
          pccB[(size_t)ig * ROIS_ + rg] = v;
        }
      }
    }
  }
}

// ---------------------------------------------------------------------------
// Kernel 3: e[b,o,r] = sum_k pcc[b,r,k]*ec_w[o,k] + ec_b[o]; xe = leaky(e);
//           s[b,r] = mean_o xe
// ---------------------------------------------------------------------------
__global__ __launch_bounds__(64)
void fsipm_edge_kernel(const float* __restrict__ PCC,
                       const float* __restrict__ ecw, const float* __restrict__ ecb,
                       float* __restrict__ XE, float* __restrict__ S) {
  const int r = blockIdx.x, b = blockIdx.y, tid = threadIdx.x;
  const float* row = PCC + ((size_t)b * ROIS_ + r) * ROIS_;
  float acc[C1_] = {};
  for (int k = tid; k < ROIS_; k += 64) {
    const float p = row[k];
#pragma unroll
    for (int o = 0; o < C1_; ++o) acc[o] += p * ecw[o * ROIS_ + k];
  }
  __shared__ float red[C1_][64];
  __shared__ float xeSh[C1_];
#pragma unroll
  for (int o = 0; o < C1_; ++o) red[o][tid] = acc[o];
  __syncthreads();
  if (tid < C1_) {
    float s = 0.f;
    for (int i = 0; i < 64; ++i) s += red[tid][i];
    const float xe = leakyf(s + ecb[tid]);
    XE[((size_t)b * ROIS_ + r) * C1_ + tid] = xe;
    xeSh[tid] = xe;
  }
  __syncthreads();
  if (tid == 0) {
    float s = 0.f;
#pragma unroll
    for (int o = 0; o < C1_; ++o) s += xeSh[o];
    S[b * ROIS_ + r] = s * (1.0f / C1_);
  }
}

// ---------------------------------------------------------------------------
// Kernel 4: per-batch MLP 400 -> relu(100) -> 400 -> sigmoid = att.
// Writes att straight into d_out (att4 region) and to ws for kernel 5.
// ---------------------------------------------------------------------------
__global__ __launch_bounds__(128)
void fsipm_mlp_kernel(const float* __restrict__ S,
                      const float* __restrict__ w1, const float* __restrict__ b1,
                      const float* __restrict__ w2, const float* __restrict__ b2,
                      float* __restrict__ ATT, float* __restrict__ out_att) {
  const int b = blockIdx.x, tid = threadIdx.x;
  __shared__ float sS[ROIS_], sH[HID_];
  for (int r = tid; r < ROIS_; r += 128) sS[r] = S[b * ROIS_ + r];
  __syncthreads();
  if (tid < HID_) {
    float a = b1[tid];
    for (int r = 0; r < ROIS_; ++r) a += w1[tid * ROIS_ + r] * sS[r];
    sH[tid] = fmaxf(a, 0.f);
  }
  __syncthreads();
  for (int r = tid; r < ROIS_; r += 128) {
    float a = b2[r];
    for (int j = 0; j < HID_; ++j) a += w2[r * HID_ + j] * sH[j];
    const float att = 1.f / (1.f + __expf(-a));
    ATT[b * ROIS_ + r] = att;
    out_att[b * ROIS_ + r] = att;
  }
}

// ---------------------------------------------------------------------------
// Kernel 5: g[b,o] = leaky( sum_{r,c} att[b,r]*xe[b,r,c]*nc_w[o,c,r] + nc_b[o] )
// LDS ds_add_f32 atomics for the cross-thread reduction.
// ---------------------------------------------------------------------------
__global__ __launch_bounds__(256)
void fsipm_head_kernel(const float* __restrict__ ATT, const float* __restrict__ XE,
                       const float* __restrict__ ncw, const float* __restrict__ ncb,
                       float* __restrict__ out_g) {
  const int b = blockIdx.x, tid = threadIdx.x;
  __shared__ float g[C2_];
  if (tid < C2_) g[tid] = 0.f;
  __syncthreads();
  float part[C2_] = {};
  for (int r = tid; r < ROIS_; r += 256) {
    const float a = ATT[b * ROIS_ + r];
#pragma unroll
    for (int c = 0; c < C1_; ++c) {
      const float v = a * XE[((size_t)b * ROIS_ + r) * C1_ + c];
#pragma unroll
      for (int o = 0; o < C2_; ++o) part[o] += v * ncw[(o * C1_ + c) * ROIS_ + r];
    }
  }
#pragma unroll
  for (int o = 0; o < C2_; ++o) atomicAdd(&g[o], part[o]);
  __syncthreads();
  if (tid < C2_) out_g[b * C2_ + tid] = leakyf(g[tid] + ncb[tid]);
}

// ---------------------------------------------------------------------------
extern "C" void kernel_launch(void* const* d_in, const int* in_sizes, int n_in,
                              void* d_out, int out_size, void* d_ws, size_t ws_size,
                              hipStream_t stream) {
  const float* x   = (const float*)d_in[0];
  const float* ecw = (const float*)d_in[1];
  const float* ecb = (const float*)d_in[2];
  const float* w1  = (const float*)d_in[3];
  const float* b1  = (const float*)d_in[4];
  const float* w2  = (const float*)d_in[5];
  const float* b2  = (const float*)d_in[6];
  const float* ncw = (const float*)d_in[7];
  const float* ncb = (const float*)d_in[8];
  float* out = (float*)d_out;  // [0,1024): g ; [1024, 1024+25600): att4

  // Workspace carve-up (floats): ~42.2 MB total
  float* ws   = (float*)d_ws;
  float* PCC  = ws;                                   // B*400*400 = 10,240,000
  float* MU   = PCC + (size_t)B_ * ROIS_ * ROIS_;     // B*400
  float* INVN = MU + B_ * ROIS_;                      // B*400
  float* XE   = INVN + B_ * ROIS_;                    // B*400*8
  float* S    = XE + (size_t)B_ * ROIS_ * C1_;        // B*400
  float* ATT  = S + B_ * ROIS_;                       // B*400

  fsipm_stats_kernel<<<dim3(ROIS_, B_), 64, 0, stream>>>(x, MU, INVN);
  fsipm_gram_pcc_kernel<<<dim3(NPAIR_, B_), 128, 0, stream>>>(x, MU, INVN, PCC);
  fsipm_edge_kernel<<<dim3(ROIS_, B_), 64, 0, stream>>>(PCC, ecw, ecb, XE, S);
  fsipm_mlp_kernel<<<B_, 128, 0, stream>>>(S, w1, b1, w2, b2, ATT, out + B_ * C2_);
  fsipm_head_kernel<<<B_, 256, 0, stream>>>(ATT, XE, ncw, ncb, out);
}